// SampleAndGroup_34102040330298
// MI455X (gfx1250) — compile-verified
//
#include <hip/hip_runtime.h>

#define B_  8
#define N_  8192
#define C_  128
#define S_  1024
#define NS_ 32
#define R2_ (0.2f * 0.2f)

typedef __attribute__((ext_vector_type(2))) float v2f;
typedef __attribute__((ext_vector_type(8))) float v8f;

// ---------------------------------------------------------------------------
// Kernel 1: farthest point sampling. One 1024-thread block (32 waves) per
// batch. Each thread owns 8 points in registers; per step: distance update,
// register argmax, wave32 shuffle reduce, 32-partial LDS reduce, centroid
// broadcast. Writes sample_xyz directly to d_out.
// ---------------------------------------------------------------------------
__global__ __launch_bounds__(1024) void fps_kernel(const float* __restrict__ xyz,
                                                   float* __restrict__ out_xyz)
{
    const int b = blockIdx.x;
    const int t = threadIdx.x;
    const float* bx = xyz + (size_t)b * N_ * 3;

    float px[8], py[8], pz[8], dist[8];
#pragma unroll
    for (int k = 0; k < 8; ++k) {
        int p = t + k * 1024;
        px[k] = bx[p * 3 + 0];
        py[k] = bx[p * 3 + 1];
        pz[k] = bx[p * 3 + 2];
        dist[k] = 1e10f;
    }

    __shared__ float s_val[32];
    __shared__ int   s_idx[32];
    __shared__ float s_cent[3];
    __shared__ int   s_far;

    if (t == 0) { s_cent[0] = px[0]; s_cent[1] = py[0]; s_cent[2] = pz[0]; s_far = 0; }
    __syncthreads();

    for (int s = 0; s < S_; ++s) {
        float cx = s_cent[0], cy = s_cent[1], cz = s_cent[2];
        if (t == 0) {
            float* o = out_xyz + ((size_t)b * S_ + s) * 3;
            o[0] = cx; o[1] = cy; o[2] = cz;
        }
        float bv = -1.0f; int bi = 0;
#pragma unroll
        for (int k = 0; k < 8; ++k) {
            float dx = px[k] - cx, dy = py[k] - cy, dz = pz[k] - cz;
            float d = dx * dx + dy * dy + dz * dz;
            dist[k] = fminf(dist[k], d);
            if (dist[k] > bv) { bv = dist[k]; bi = t + k * 1024; }
        }
        // wave32 argmax reduce, tie -> lowest global index (matches jnp.argmax)
#pragma unroll
        for (int off = 16; off > 0; off >>= 1) {
            float ov = __shfl_xor(bv, off, 32);
            int   oi = __shfl_xor(bi, off, 32);
            if (ov > bv || (ov == bv && oi < bi)) { bv = ov; bi = oi; }
        }
        if ((t & 31) == 0) { s_val[t >> 5] = bv; s_idx[t >> 5] = bi; }
        __syncthreads();
        if (t < 32) {
            bv = s_val[t]; bi = s_idx[t];
#pragma unroll
            for (int off = 16; off > 0; off >>= 1) {
                float ov = __shfl_xor(bv, off, 32);
                int   oi = __shfl_xor(bi, off, 32);
                if (ov > bv || (ov == bv && oi < bi)) { bv = ov; bi = oi; }
            }
            if (t == 0) s_far = bi;
        }
        __syncthreads();
        int far = s_far;
        if (t == (far & 1023)) {
            int k = far >> 10;
            s_cent[0] = px[k]; s_cent[1] = py[k]; s_cent[2] = pz[k];
        }
        __syncthreads();
    }
}

// ---------------------------------------------------------------------------
// Kernel 2: ball query (WMMA f32 16x16x4 pairwise d^2) + ordered radius
// selection + fused feature gather. One 256-thread block per (batch, tile of
// 16 sample points).
//   d^2[m][n] = A(m,:)·B(:,n) + C[m][n]
//   A row m   = (-2sx, -2sy, -2sz, |s|^2)   (16x4, 2 VGPRs/lane)
//   B col n   = ( px,   py,   pz,  1    )   (4x16, 2 VGPRs/lane)
//   C[m][n]   = |x_n|^2  (row-invariant => every D VGPR seeded with xn2[lane%16])
// The D lane layout makes one __ballot per D VGPR v yield rows v (bits 15:0)
// and v+8 (bits 31:16) of the tile. All 8 ballot words for a chunk are
// gathered in registers and lane 0 writes them with two ds_store_b128 into a
// chunk-major u32[512][8] LDS array (one EXEC branch per chunk instead of 8).
// ---------------------------------------------------------------------------
__global__ __launch_bounds__(256) void ballquery_gather_kernel(
    const float* __restrict__ xyz, const float* __restrict__ feat,
    const float* __restrict__ sxyz, float* __restrict__ outg)
{
    const int tile = blockIdx.x & 63;
    const int b    = blockIdx.x >> 6;
    const int tid  = threadIdx.x;
    const int wave = tid >> 5, lane = tid & 31;
    const int n16  = lane & 15, h = lane >> 4;

    __shared__ unsigned s_maskT[512][8];   // [chunk][v]: rows v (lo16) & v+8 (hi16)
    __shared__ int      s_nidx[16][NS_];

    // Loop-invariant A operand for this lane (row m = n16, K = 2h, 2h+1).
    const int sid = tile * 16 + n16;
    const float* sp = sxyz + ((size_t)b * S_ + sid) * 3;
    const float sx = sp[0], sy = sp[1], sz = sp[2];
    const float sn2 = sx * sx + sy * sy + sz * sz;
    v2f A;
    A.x = h ? (-2.0f * sz) : (-2.0f * sx);
    A.y = h ? sn2          : (-2.0f * sy);

    const float* bx = xyz + (size_t)b * N_ * 3;

#pragma unroll 2
    for (int chunk = wave; chunk < 512; chunk += 8) {
        const int p = chunk * 16 + n16;
        const float qx = bx[p * 3 + 0], qy = bx[p * 3 + 1], qz = bx[p * 3 + 2];
        const float xn2 = qx * qx + qy * qy + qz * qz;
        v2f Bm;
        Bm.x = h ? qz   : qx;
        Bm.y = h ? 1.0f : qy;
        v8f Cm;
#pragma unroll
        for (int i = 0; i < 8; ++i) Cm[i] = xn2;

        v8f D = __builtin_amdgcn_wmma_f32_16x16x4_f32(
            /*neg_a=*/false, A, /*neg_b=*/false, Bm,
            /*c_mod=*/(short)0, Cm, /*reuse_a=*/false, /*reuse_b=*/false);

        // Wave-uniform ballot words for all 8 row pairs of this 16x16 tile.
        unsigned bal[8];
#pragma unroll
        for (int v = 0; v < 8; ++v)
            bal[v] = (unsigned)__ballot(D[v] <= R2_);

        if (lane == 0) {
            uint4* dst = (uint4*)&s_maskT[chunk][0];
            dst[0] = make_uint4(bal[0], bal[1], bal[2], bal[3]);
            dst[1] = make_uint4(bal[4], bal[5], bal[6], bal[7]);
        }
    }
    __syncthreads();

    // Ordered selection: first NS_ in-radius indices, ascending; pad w/ first.
    if (tid < 16) {
        const int m    = tid;
        const int vcol = m & 7;
        const int sh   = (m >> 3) << 4;   // rows 8..15 live in bits 31:16
        int cnt = 0;
        for (int c = 0; c < 512 && cnt < NS_; ++c) {
            unsigned w = (s_maskT[c][vcol] >> sh) & 0xFFFFu;
            while (w && cnt < NS_) {
                int bpos = __ffs(w) - 1;
                s_nidx[m][cnt++] = c * 16 + bpos;
                w &= (w - 1);
            }
        }
        const int first = (cnt > 0) ? s_nidx[m][0] : 0;
        for (; cnt < NS_; ++cnt) s_nidx[m][cnt] = first;
    }
    __syncthreads();

    // Fused gather: 16*32 = 512 rows of 128 f32; one float4 per lane per row.
    for (int r = wave; r < 512; r += 8) {
        const int m = r >> 5, k = r & 31;
        const int src = s_nidx[m][k];
        const float4* fs = (const float4*)(feat + ((size_t)b * N_ + src) * C_);
        float4* fd = (float4*)(outg +
            ((((size_t)b * S_ + tile * 16 + m) * NS_) + k) * C_);
        fd[lane] = fs[lane];
    }
}

extern "C" void kernel_launch(void* const* d_in, const int* in_sizes, int n_in,
                              void* d_out, int out_size, void* d_ws, size_t ws_size,
                              hipStream_t stream)
{
    (void)in_sizes; (void)n_in; (void)out_size; (void)d_ws; (void)ws_size;

    const float* xyz  = (const float*)d_in[0];   // (8, 8192, 3) f32
    const float* feat = (const float*)d_in[1];   // (8, 8192, 128) f32

    float* out      = (float*)d_out;
    float* out_xyz  = out;                               // (8, 1024, 3)
    float* out_grp  = out + (size_t)B_ * S_ * 3;         // (8, 1024, 32, 128)

    fps_kernel<<<B_, 1024, 0, stream>>>(xyz, out_xyz);
    ballquery_gather_kernel<<<B_ * 64, 256, 0, stream>>>(xyz, feat, out_xyz, out_grp);
}